// HistToPointCloud_43559558316078
// MI455X (gfx1250) — compile-verified
//
#include <hip/hip_runtime.h>
#include <hip/hip_bf16.h>

// HistToPointCloud: per-batch stable stream compaction of nonzero histogram
// bins into a point cloud. Memory-bound: ~265 MB of DRAM traffic total
// (hist 64MB read [2nd read L2-resident: 192MB global L2 >> 64MB hist],
// outputs ~192MB written) -> ~11-14us floor at 23.3 TB/s on MI455X.
// No matmul structure -> WMMA inapplicable; the CDNA5 win is data movement:
//  * pass 1 uses the Tensor Data Mover (tensor_load_to_lds + s_wait_tensorcnt)
//    to DMA each 16KB hist tile into LDS (order-irrelevant counting), with a
//    conflict-free interleaved LDS read pattern (lane L -> bank L).
//  * pass 3 streams b128 loads from L2 and writes every output slot exactly
//    once (stable-partition tail slots get their zeros directly -> no
//    separate 200MB zero-fill pass).

#define BSZ   64
#define DXD   512
#define DYD   512
#define LTOT  (DXD * DYD)        // 262144 elements per batch
#define TILE  4096               // elements per block (16 KB)
#define NT    (LTOT / TILE)      // 64 tiles per batch
#define BLOCK 256                // 8 wave32 waves
#define EPT   (TILE / BLOCK)     // 16 elements per thread

#if defined(__gfx1250__) && __has_builtin(__builtin_amdgcn_tensor_load_to_lds) && \
    __has_builtin(__builtin_amdgcn_s_wait_tensorcnt)
#define USE_TDM 1
#else
#define USE_TDM 0
#endif

typedef unsigned int u32x4 __attribute__((ext_vector_type(4)));
typedef int          i32x8 __attribute__((ext_vector_type(8)));
typedef int          i32x4 __attribute__((ext_vector_type(4)));

#if USE_TDM
// Toolchain-portable TDM issue: clang-23 (amdgpu-toolchain/therock) takes the
// 6-arg form (extra int32x8 group, zero-filled); clang-22 (ROCm 7.2) takes 5.
__device__ __forceinline__ void tdm_load_to_lds(u32x4 g0, i32x8 g1) {
    i32x4 z4 = { 0, 0, 0, 0 };            // groups 2/3 unused (<=2D tensor)
#if __clang_major__ >= 23
    i32x8 z8 = { 0, 0, 0, 0, 0, 0, 0, 0 };
    __builtin_amdgcn_tensor_load_to_lds(g0, g1, z4, z4, z8, 0);
#else
    __builtin_amdgcn_tensor_load_to_lds(g0, g1, z4, z4, 0);
#endif
}
#endif

// ---------------------------------------------------------------- pass 1
// Per-tile nonzero count.
__global__ void k_count(const float* __restrict__ hist, int* __restrict__ tileCnt) {
    const int t = blockIdx.x, b = blockIdx.y, tid = threadIdx.x;
    int cnt = 0;
#if USE_TDM
    __shared__ float sbuf[TILE];  // 16 KB of the 320 KB/WGP LDS
    if (tid < 32) {  // one wave issues the DMA (TDM ignores EXEC; waves 1-7 branch over it)
        const unsigned long long ga =
            (unsigned long long)(uintptr_t)(hist + (size_t)b * LTOT + (size_t)t * TILE);
        const unsigned int ga_lo = (unsigned int)ga;
        const unsigned int ga_hi = (unsigned int)(ga >> 32);
        const unsigned int lds_a = (unsigned int)(uintptr_t)&sbuf[0];  // low 32b = LDS offset
        // D# group 0: [1:0]=count=1 | lds_addr | global_addr[56:0] | type=2 (bits 127:126)
        u32x4 g0 = { 1u, lds_a, ga_lo, (ga_hi & 0x01FFFFFFu) | 0x80000000u };
        // D# group 1 (bit-packed per ISA 8.4):
        //  w0: data_size=2 (4B) in [17:16]; mask/pad/iterate = 0
        //  w1: tensor_dim0[15:0]=4096 in [31:16]
        //  w2: tensor_dim0[31:16]=0 | tensor_dim1[15:0]=1 in [31:16]
        //  w3: tensor_dim1[31:16]=0 | tile_dim0=4096 in [31:16]
        //  w4: tile_dim1=1 | tile_dim2=0
        //  w5: tensor_dim0_stride[31:0]=4096 ; w6,w7: high stride bits = 0
        i32x8 g1 = { 0x20000, 0x10000000, 0x10000, 0x10000000, 1, 4096, 0, 0 };
        tdm_load_to_lds(g0, g1);
        __builtin_amdgcn_s_wait_tensorcnt(0);  // TENSORcnt==0: tile is in LDS
    }
    __syncthreads();
    // Interleaved reads: lane L always hits LDS bank L -> conflict-free.
#pragma unroll
    for (int q = 0; q < EPT; ++q) cnt += (sbuf[tid + BLOCK * q] != 0.0f);
#else
    const size_t base = (size_t)b * LTOT + (size_t)t * TILE + (size_t)tid * EPT;
    const float4* hp = (const float4*)(hist + base);
#pragma unroll
    for (int q = 0; q < 4; ++q) {
        float4 v = hp[q];
        cnt += (v.x != 0.0f) + (v.y != 0.0f) + (v.z != 0.0f) + (v.w != 0.0f);
    }
#endif
    __shared__ int sm[BLOCK];
    sm[tid] = cnt;
    __syncthreads();
    for (int o = BLOCK / 2; o > 0; o >>= 1) {
        if (tid < o) sm[tid] += sm[tid + o];
        __syncthreads();
    }
    if (tid == 0) tileCnt[b * NT + t] = sm[0];
}

// ---------------------------------------------------------------- pass 2
// Per-batch exclusive scan of NT=64 tile counts; emit lens (ws + d_out).
__global__ void k_scan(const int* __restrict__ tileCnt, int* __restrict__ tileOff,
                       int* __restrict__ lens, int* __restrict__ lens_out) {
    const int b = blockIdx.x, tid = threadIdx.x;  // 64 threads = 2 waves
    __shared__ int sm[NT];
    const int c = tileCnt[b * NT + tid];
    sm[tid] = c;
    __syncthreads();
    for (int o = 1; o < NT; o <<= 1) {            // inclusive Hillis-Steele
        int v = (tid >= o) ? sm[tid - o] : 0;
        __syncthreads();
        sm[tid] += v;
        __syncthreads();
    }
    tileOff[b * NT + tid] = sm[tid] - c;          // exclusive offset
    if (tid == NT - 1) {
        lens[b]     = sm[tid];                    // for pass 3
        lens_out[b] = sm[tid];                    // int32 bits into d_out
    }
}

// ---------------------------------------------------------------- pass 3
// Intra-tile stable rank + scatter. Nonzero element -> tileOff + local rank.
// Zero element -> lens + (flat_idx - nonzero_prefix)  (its stable tail slot,
// value 0) => every output position in [0, L) written exactly once.
__global__ void k_compact(const float* __restrict__ hist,
                          const float* __restrict__ xl,
                          const float* __restrict__ yl,
                          const int* __restrict__ tileOff,
                          const int* __restrict__ lens,
                          float* __restrict__ pc,    // [BS, L, 2]
                          float* __restrict__ pcw) { // [BS, L]
    const int t = blockIdx.x, b = blockIdx.y, tid = threadIdx.x;
    const int flat0 = t * TILE + tid * EPT;           // batch-local flat index
    const size_t base = (size_t)b * LTOT + flat0;

    // gfx1250 global_prefetch_b8: kick the (L2-resident) tile fetch early.
    __builtin_prefetch(hist + base, 0, 1);

    float v[EPT];
    const float4* hp = (const float4*)(hist + base);  // L2 hit (warmed in pass 1)
#pragma unroll
    for (int q = 0; q < 4; ++q) {
        float4 a = hp[q];
        v[4 * q + 0] = a.x; v[4 * q + 1] = a.y;
        v[4 * q + 2] = a.z; v[4 * q + 3] = a.w;
    }
    int cnt = 0;
#pragma unroll
    for (int k = 0; k < EPT; ++k) cnt += (v[k] != 0.0f);

    __shared__ int sm[BLOCK];
    sm[tid] = cnt;
    __syncthreads();
    for (int o = 1; o < BLOCK; o <<= 1) {             // inclusive scan
        int x = (tid >= o) ? sm[tid - o] : 0;
        __syncthreads();
        sm[tid] += x;
        __syncthreads();
    }
    const int excl = sm[tid] - cnt;

    int nz = tileOff[b * NT + t] + excl;  // running global nonzero rank
    const int len = lens[b];

    // cx[i] = lin[i]*cxw*(dx-1) + x0 + cxw/2  ==  x0 + (i+0.5)*cxw
    const float x0 = xl[2 * b], cxw = (xl[2 * b + 1] - x0) * (1.0f / DXD);
    const float y0 = yl[2 * b], cyw = (yl[2 * b + 1] - y0) * (1.0f / DYD);
    const float xoff = x0 + 0.5f * cxw;
    const float yoff = y0 + 0.5f * cyw;

    float* pcb = pc + (size_t)b * LTOT * 2;
    float* pwb = pcw + (size_t)b * LTOT;

#pragma unroll
    for (int k = 0; k < EPT; ++k) {
        const int fi = flat0 + k;
        float w = v[k];
        float px, py, pw;
        int pos;
        if (w != 0.0f) {
            const float hc = fminf(fmaxf(w, 0.0f), 1.0f);
            const int i = fi >> 9;           // fi / DY
            const int j = fi & (DYD - 1);    // fi % DY
            px = fmaf((float)i, cxw, xoff) * hc;
            py = fmaf((float)j, cyw, yoff) * hc;
            pw = w;
            pos = nz++;
        } else {
            px = 0.0f; py = 0.0f; pw = 0.0f;
            pos = len + (fi - nz);           // stable tail slot
        }
        pcb[2 * (size_t)pos + 0] = px;
        pcb[2 * (size_t)pos + 1] = py;
        pwb[pos] = pw;
    }
}

extern "C" void kernel_launch(void* const* d_in, const int* in_sizes, int n_in,
                              void* d_out, int out_size, void* d_ws, size_t ws_size,
                              hipStream_t stream) {
    const float* hist = (const float*)d_in[0];   // [64, 512, 512] f32
    const float* xl   = (const float*)d_in[1];   // [64, 2] f32
    const float* yl   = (const float*)d_in[2];   // [64, 2] f32

    float* out      = (float*)d_out;
    float* pc       = out;                              // BS*L*2 floats
    float* pcw      = out + (size_t)BSZ * LTOT * 2;     // BS*L floats
    int*   lens_out = (int*)(out + (size_t)BSZ * LTOT * 3);  // BS int32 (bitwise)

    int* ws      = (int*)d_ws;      // needs 2*BS*NT + BS ints (~33 KB)
    int* tileCnt = ws;              // [BS*NT]
    int* tileOff = ws + BSZ * NT;   // [BS*NT]
    int* lens    = ws + 2 * BSZ * NT;  // [BS]

    dim3 grid(NT, BSZ);
    k_count  <<<grid, BLOCK, 0, stream>>>(hist, tileCnt);
    k_scan   <<<BSZ, NT, 0, stream>>>(tileCnt, tileOff, lens, lens_out);
    k_compact<<<grid, BLOCK, 0, stream>>>(hist, xl, yl, tileOff, lens, pc, pcw);
}